// Net_22608707846799
// MI455X (gfx1250) — compile-verified
//
#include <hip/hip_runtime.h>

typedef __attribute__((ext_vector_type(2))) float v2f;
typedef __attribute__((ext_vector_type(8))) float v8f;

#define N_VEC   100   // n
#define N_PAD   112   // padded to 7 tiles of 16
#define N_TILES 7
#define KB      64    // K rows staged per LDS chunk
#define GRAM_BLOCK 224 // 7 waves (wave32): one wave per 16-row tile strip
#define GRAM_GRID  512
#define K_SEL   49    // k+1 = n - n/2 - 2 + 1

// ---------------------------------------------------------------- zero G
__global__ void zero_g(float* G) {
    int i = blockIdx.x * blockDim.x + threadIdx.x;
    if (i < N_PAD * N_PAD) G[i] = 0.0f;
}

// ---------------------------------------------------------------- Gram: G = A^T A
// A is the [D, 100] view of the input (100 contiguous per d).
// Each workgroup: stage KB x 112 fp32 rows into LDS, then wave w accumulates
// the full tile-row (w, 0..6) of the 7x7 tile grid via V_WMMA_F32_16X16X4_F32.
// Partial 112x112 Gram reduced with native f32 add atomics (no-return).
__global__ __launch_bounds__(GRAM_BLOCK)
void gram_kernel(const float* __restrict__ x, float* __restrict__ G, int D) {
    __shared__ float xs[KB * N_PAD];
    const int tid  = threadIdx.x;
    const int wv   = tid >> 5;      // wave id == tile row ti (0..6), uniform per wave
    const int lane = tid & 31;
    const int half = lane >> 4;     // lane half selects K pair (ISA A/B f32 layout)
    const int mn   = lane & 15;     // M (for A) / N (for B,C,D)

    v8f acc[N_TILES];
#pragma unroll
    for (int t = 0; t < N_TILES; ++t) acc[t] = (v8f){0,0,0,0,0,0,0,0};

    for (int d0 = blockIdx.x * KB; d0 < D; d0 += gridDim.x * KB) {
        int rv = D - d0; if (rv > KB) rv = KB;
        // stage KB x 100 (zero-padded to 112) into LDS
        for (int i = tid; i < KB * N_PAD; i += GRAM_BLOCK) {
            int r = i / N_PAD, c = i - r * N_PAD;
            float v = 0.0f;
            if (r < rv && c < N_VEC) v = x[(size_t)(d0 + r) * N_VEC + c];
            xs[i] = v;
        }
        __syncthreads();

        for (int kk = 0; kk < KB; kk += 4) {
            // A fragment, 16x4 f32: lanes 0-15 -> K = kk,kk+1 ; lanes 16-31 -> K = kk+2,kk+3
            const int base = (kk + 2 * half) * N_PAD + mn;
            v2f a;
            a.x = xs[base + 16 * wv];
            a.y = xs[base + N_PAD + 16 * wv];
#pragma unroll
            for (int tj = 0; tj < N_TILES; ++tj) {
                v2f b;                                  // B 4x16 f32: mirror layout
                b.x = xs[base + 16 * tj];
                b.y = xs[base + N_PAD + 16 * tj];
                acc[tj] = __builtin_amdgcn_wmma_f32_16x16x4_f32(
                    false, a, false, b, (short)0, acc[tj], false, false);
            }
        }
        __syncthreads();
    }

    // C/D 16x16 f32 layout: lane -> N (mod 16), VGPR r + 8*half -> M
#pragma unroll
    for (int tj = 0; tj < N_TILES; ++tj) {
#pragma unroll
        for (int r = 0; r < 8; ++r) {
            int row = wv * 16 + r + 8 * half;
            int col = tj * 16 + mn;
            unsafeAtomicAdd(&G[row * N_PAD + col], acc[tj][r]);
        }
    }
}

// ---------------------------------------------------------------- Krum selection
// dist[i][j] = sqrt(max(G[ii]+G[jj]-2G[ij],0)); per row pick 49 smallest
// (tie -> lowest index, matching top_k order), sum -> score; argmin row -> idx list.
__global__ void krum_select(const float* __restrict__ G, int* __restrict__ idx_out) {
    __shared__ float dist[N_VEC][N_VEC];   // 40000 B
    __shared__ float score[N_VEC];
    __shared__ int   sel[N_VEC][K_SEL];    // 19600 B
    const int tid = threadIdx.x;

    for (int e = tid; e < N_VEC * N_VEC; e += blockDim.x) {
        int i = e / N_VEC, j = e - i * N_VEC;
        float gii = G[i * N_PAD + i];
        float gjj = G[j * N_PAD + j];
        float gij = G[i * N_PAD + j];
        float d2  = fmaxf(gii + gjj - 2.0f * gij, 0.0f);
        dist[i][j] = sqrtf(d2);
    }
    __syncthreads();

    if (tid < N_VEC) {
        float s = 0.0f;
        for (int t = 0; t < K_SEL; ++t) {
            float best = 3.4e38f; int bi = 0;
            for (int j = 0; j < N_VEC; ++j) {
                float v = dist[tid][j];
                if (v < best) { best = v; bi = j; }
            }
            s += best;
            sel[tid][t] = bi;
            dist[tid][bi] = 3.4e38f;
        }
        score[tid] = s;
    }
    __syncthreads();

    if (tid == 0) {
        float best = 3.4e38f; int bi = 0;
        for (int i = 0; i < N_VEC; ++i)
            if (score[i] < best) { best = score[i]; bi = i; }
        for (int t = 0; t < K_SEL; ++t) idx_out[t] = sel[bi][t];
    }
}

// ---------------------------------------------------------------- mean over selected columns
__global__ void mkrum_mean(const float* __restrict__ x, const int* __restrict__ idx,
                           float* __restrict__ out, int D) {
    __shared__ int s_idx[K_SEL];
    if (threadIdx.x < K_SEL) s_idx[threadIdx.x] = idx[threadIdx.x];
    __syncthreads();
    for (int d = blockIdx.x * blockDim.x + threadIdx.x; d < D;
         d += gridDim.x * blockDim.x) {
        const float* row = x + (size_t)d * N_VEC;
        float s = 0.0f;
#pragma unroll
        for (int t = 0; t < K_SEL; ++t) s += row[s_idx[t]];
        out[d] = s * (1.0f / (float)K_SEL);
    }
}

// ---------------------------------------------------------------- launch
extern "C" void kernel_launch(void* const* d_in, const int* in_sizes, int n_in,
                              void* d_out, int out_size, void* d_ws, size_t ws_size,
                              hipStream_t stream) {
    const float* x = (const float*)d_in[0];
    float* out = (float*)d_out;
    const int D = in_sizes[0] / N_VEC;          // 1,000,000

    float* G   = (float*)d_ws;                                  // 112*112 fp32
    int*   idx = (int*)((char*)d_ws + N_PAD * N_PAD * sizeof(float)); // 49 ints

    zero_g<<<(N_PAD * N_PAD + 255) / 256, 256, 0, stream>>>(G);
    gram_kernel<<<GRAM_GRID, GRAM_BLOCK, 0, stream>>>(x, G, D);
    krum_select<<<1, 128, 0, stream>>>(G, idx);
    mkrum_mean<<<2048, 256, 0, stream>>>(x, idx, out, D);
}